// DemeanedGlobalLossD_10557029614244
// MI455X (gfx1250) — compile-verified
//
#include <hip/hip_runtime.h>
#include <math.h>

#define NROWS 384
#define BSZ   192
#define PD    16
#define DLEN  73728            // 32*48*48
#define TEMP_INV 10.0f
#define EPSN  1e-6f

#define KCHUNKS 8
#define KCHUNK  (DLEN / KCHUNKS)   // 9216, divisible by 32
#define ITILES  24                 // 384/16
#define JGROUPS 6                  // 24 tiles / 4 per wave
#define JPER    4
#define TASKS   (ITILES * JGROUPS * KCHUNKS)   // 1152 waves
#define GRAM_BLOCKS (TASKS / 8)                // 8 waves (256 thr) per block

typedef __attribute__((ext_vector_type(2))) float v2f;
typedef __attribute__((ext_vector_type(8))) float v8f;

__device__ __forceinline__ int ds_of_row(int r) { return (r >> 2) & 3; }

// ---------------------------------------------------------------- norms
__global__ void dgl_norms(const float* __restrict__ X, const float* __restrict__ M,
                          float* __restrict__ inv_nrm) {
  __shared__ float red[256];
  const int n = blockIdx.x;
  const float* xr = X + (size_t)n * DLEN;
  const float* mr = M + (size_t)ds_of_row(n) * DLEN;
  float acc = 0.f;
  for (int d = threadIdx.x; d < DLEN; d += 256) {
    float z = xr[d] - mr[d];
    acc += z * z;
  }
  red[threadIdx.x] = acc;
  __syncthreads();
  for (int s = 128; s > 0; s >>= 1) {
    if ((int)threadIdx.x < s) red[threadIdx.x] += red[threadIdx.x + s];
    __syncthreads();
  }
  if (threadIdx.x == 0) inv_nrm[n] = 1.0f / fmaxf(sqrtf(red[0]), EPSN);
}

// ---------------------------------------------------------------- zero scratch
__global__ void dgl_zero(float* __restrict__ p, int n) {
  int i = blockIdx.x * blockDim.x + threadIdx.x;
  if (i < n) p[i] = 0.f;
}

// ---------------------------------------------------------------- gram (WMMA fp32)
__global__ void dgl_gram(const float* __restrict__ X, const float* __restrict__ M,
                         float* __restrict__ Sraw) {
  const int lane = threadIdx.x & 31;
  const int wave = threadIdx.x >> 5;
  const int t    = blockIdx.x * 8 + wave;          // 0..TASKS-1
  const int I    = t / (JGROUPS * KCHUNKS);
  const int rm   = t % (JGROUPS * KCHUNKS);
  const int Jg   = rm / KCHUNKS;
  const int kc   = rm % KCHUNKS;

  // A-matrix 16x4 f32 layout: lanes 0-15 -> M=0..15, VGPR0/1 = K {0,1};
  // lanes 16-31 -> same M, K {2,3}. float2 load at col offset k + 2*(lane>=16).
  const int m     = lane & 15;
  const int khalf = (lane >> 4) * 2;

  const int rowA  = I * 16 + m;
  const float* xa = X + (size_t)rowA * DLEN;
  const float* ma = M + (size_t)ds_of_row(rowA) * DLEN;

  const int J0 = Jg * JPER;
  const float* xb[JPER];
  const float* mb[JPER];
#pragma unroll
  for (int j = 0; j < JPER; ++j) {
    int rowB = (J0 + j) * 16 + m;
    xb[j] = X + (size_t)rowB * DLEN;
    mb[j] = M + (size_t)ds_of_row(rowB) * DLEN;
  }

  v8f c[JPER];
#pragma unroll
  for (int j = 0; j < JPER; ++j) { v8f z = {}; c[j] = z; }

  const int k0 = kc * KCHUNK;
  const int k1 = k0 + KCHUNK;
  for (int k = k0; k < k1; k += 32) {              // 8 WMMAs of K=4 per iter
    v2f a[8];
#pragma unroll
    for (int u = 0; u < 8; ++u) {
      const int kk = k + 4 * u + khalf;
      v2f xv = *(const v2f*)(xa + kk);
      v2f mv = *(const v2f*)(ma + kk);
      a[u] = xv - mv;                              // demean on the fly
    }
#pragma unroll
    for (int j = 0; j < JPER; ++j) {               // reuse A across 4 J tiles
#pragma unroll
      for (int u = 0; u < 8; ++u) {
        const int kk = k + 4 * u + khalf;
        v2f xv = *(const v2f*)(xb[j] + kk);
        v2f mv = *(const v2f*)(mb[j] + kk);
        v2f b  = xv - mv;
        c[j] = __builtin_amdgcn_wmma_f32_16x16x4_f32(
                   /*neg_a=*/false, a[u], /*neg_b=*/false, b,
                   /*c_mod=*/(short)0, c[j], /*reuse_a=*/false, /*reuse_b=*/false);
      }
    }
  }

  // C/D layout: VGPR r -> row r + 8*(lane>=16), col lane%16. K-split partials
  // folded with no-return f32 atomics.
  const int colN   = lane & 15;
  const int rowOff = (lane >> 4) * 8;
#pragma unroll
  for (int j = 0; j < JPER; ++j) {
#pragma unroll
    for (int r = 0; r < 8; ++r) {
      const int grow = I * 16 + rowOff + r;
      const int gcol = (J0 + j) * 16 + colN;
      atomicAdd(&Sraw[grow * NROWS + gcol], c[j][r]);
    }
  }
}

// ---------------------------------------------------------------- Dsum
__global__ void dgl_dsum(const float* __restrict__ Sraw, const float* __restrict__ inv_nrm,
                         float* __restrict__ Dsum) {
  __shared__ float red[128];
  const int a    = blockIdx.x;
  const float ia = inv_nrm[a];
  const int arem = a & (PD - 1);
  float acc = 0.f;
  for (int b = threadIdx.x; b < NROWS; b += 128) {
    if ((b & (PD - 1)) != arem) {
      float s = Sraw[a * NROWS + b] * ia * inv_nrm[b] * TEMP_INV;
      acc += expf(s);
    }
  }
  red[threadIdx.x] = acc;
  __syncthreads();
  for (int s = 64; s > 0; s >>= 1) {
    if ((int)threadIdx.x < s) red[threadIdx.x] += red[threadIdx.x + s];
    __syncthreads();
  }
  if (threadIdx.x == 0) Dsum[a] = red[0];
}

// ---------------------------------------------------------------- final loss
__global__ void dgl_loss(const float* __restrict__ Sraw, const float* __restrict__ inv_nrm,
                         const float* __restrict__ Dsum, float* __restrict__ out) {
  __shared__ float red[BSZ];
  const int p = threadIdx.x;   // 0..191

  auto S = [&](int a, int b) -> float {
    return Sraw[a * NROWS + b] * inv_nrm[a] * inv_nrm[b] * TEMP_INV;
  };
  auto pl = [&](int a, int b) -> float {
    float num = S(a, b);
    float en  = expf(num);
    return -(num - logf(en + Dsum[a])) - (num - logf(en + Dsum[b]));
  };

  const int i1 = p;
  const int i2 = BSZ + p;
  const int i3 = (p + PD) % NROWS;
  const int i4 = (BSZ + p + PD) % NROWS;

  red[p] = pl(i1, i2) + pl(i3, i4) + pl(i1, i3) + pl(i2, i4);
  __syncthreads();
  if (p == 0) {
    float t = 0.f;
    for (int i = 0; i < BSZ; ++i) t += red[i];
    out[0] = t / (3.0f * (float)BSZ);
  }
}

// ---------------------------------------------------------------- launch
extern "C" void kernel_launch(void* const* d_in, const int* in_sizes, int n_in,
                              void* d_out, int out_size, void* d_ws, size_t ws_size,
                              hipStream_t stream) {
  (void)in_sizes; (void)n_in; (void)out_size; (void)ws_size;
  const float* X = (const float*)d_in[0];   // reg_pred [384, 32,48,48] fp32
  const float* M = (const float*)d_in[1];   // mean_representations [4, 32,48,48]
  float* out = (float*)d_out;

  float* w       = (float*)d_ws;
  float* inv_nrm = w;                // 384 floats
  float* Dsum    = w + 512;          // 384 floats
  float* Sraw    = w + 1024;         // 384*384 floats (~576 KB)

  dgl_norms<<<NROWS, 256, 0, stream>>>(X, M, inv_nrm);
  dgl_zero <<<(NROWS * NROWS + 255) / 256, 256, 0, stream>>>(Sraw, NROWS * NROWS);
  dgl_gram <<<GRAM_BLOCKS, 256, 0, stream>>>(X, M, Sraw);
  dgl_dsum <<<NROWS, 128, 0, stream>>>(Sraw, inv_nrm, Dsum);
  dgl_loss <<<1, BSZ, 0, stream>>>(Sraw, inv_nrm, Dsum, out);
}